// DynMoleRouterLoss_89103391523358
// MI455X (gfx1250) — compile-verified
//
#include <hip/hip_runtime.h>
#include <hip/hip_bf16.h>

#define L_LAYERS   32
#define E_EXP      16
#define TOP_P_C    0.8f
#define EPS_C      1e-5f
#define BCAST_THR  2.0f
#define AUX_COEF   0.001f
#define DYN_COEF   0.001f

typedef __attribute__((ext_vector_type(2))) float v2f;
typedef __attribute__((ext_vector_type(4))) float v4f;
typedef __attribute__((ext_vector_type(8))) float v8f;

// d_ws float layout: [0..15] tpe, [16..31] rppe, [32] entropy sum, [33] weight sum
__global__ void zero_ws_kernel(float* ws) {
    int i = threadIdx.x;
    if (i < 34) ws[i] = 0.0f;
}

// descending compare-exchange: max to lower index
#define CS(i, j) { float _a = v[i], _b = v[j]; v[i] = fmaxf(_a, _b); v[j] = fminf(_a, _b); }

__launch_bounds__(256)
__global__ void moe_router_main(const float* __restrict__ logits,
                                const int* __restrict__ attn,
                                float* __restrict__ ws,
                                int nTokens, int maskSize) {
    __shared__ float s_tpe[256 * 16];
    __shared__ float s_rppe[256 * 16];

    const int tid  = threadIdx.x;
    const int lane = tid & 31;

    float tpe[16], rppe[16];
#pragma unroll
    for (int e = 0; e < 16; ++e) { tpe[e] = 0.0f; rppe[e] = 0.0f; }
    float entAcc = 0.0f, wAcc = 0.0f;

    for (int tk = blockIdx.x * 256 + tid; tk < nTokens; tk += gridDim.x * 256) {
        const v4f* rp = (const v4f*)(logits + (size_t)tk * 16);
        v4f r0 = __builtin_nontemporal_load(rp + 0);   // global_load_b128 th:NT
        v4f r1 = __builtin_nontemporal_load(rp + 1);
        v4f r2 = __builtin_nontemporal_load(rp + 2);
        v4f r3 = __builtin_nontemporal_load(rp + 3);
        float v[16] = { r0.x, r0.y, r0.z, r0.w,  r1.x, r1.y, r1.z, r1.w,
                        r2.x, r2.y, r2.z, r2.w,  r3.x, r3.y, r3.z, r3.w };

        // softmax (max-subtracted, matches jax.nn.softmax)
        float m = v[0];
#pragma unroll
        for (int e = 1; e < 16; ++e) m = fmaxf(m, v[e]);
        float s = 0.0f;
#pragma unroll
        for (int e = 0; e < 16; ++e) { v[e] = __expf(v[e] - m); s += v[e]; }
        float inv = 1.0f / s;

        // entropy = -sum p*log(p+eps)
        float ent = 0.0f;
#pragma unroll
        for (int e = 0; e < 16; ++e) { v[e] *= inv; ent -= v[e] * __logf(v[e] + EPS_C); }

        // Batcher odd-even mergesort, 16 inputs, descending (63 comparators, depth 10)
        CS(0,1) CS(2,3) CS(4,5) CS(6,7) CS(8,9) CS(10,11) CS(12,13) CS(14,15)
        CS(0,2) CS(1,3) CS(4,6) CS(5,7) CS(8,10) CS(9,11) CS(12,14) CS(13,15)
        CS(1,2) CS(5,6) CS(9,10) CS(13,14)
        CS(0,4) CS(1,5) CS(2,6) CS(3,7) CS(8,12) CS(9,13) CS(10,14) CS(11,15)
        CS(2,4) CS(3,5) CS(10,12) CS(11,13)
        CS(1,2) CS(3,4) CS(5,6) CS(9,10) CS(11,12) CS(13,14)
        CS(0,8) CS(1,9) CS(2,10) CS(3,11) CS(4,12) CS(5,13) CS(6,14) CS(7,15)
        CS(4,8) CS(5,9) CS(6,10) CS(7,11)
        CS(2,4) CS(3,5) CS(6,8) CS(7,9) CS(10,12) CS(11,13)
        CS(1,2) CS(3,4) CS(5,6) CS(7,8) CS(9,10) CS(11,12) CS(13,14)

        // top-p keep mask: position e kept iff prefix-sum-before <= TOP_P, or broadcast row
        int   n  = tk % maskSize;                 // B*S token column
        float w  = (float)attn[n];
        bool  bc = (ent >= BCAST_THR);
        float cum = 0.0f;
#pragma unroll
        for (int e = 0; e < 16; ++e) {
            float kept = (bc || cum <= TOP_P_C) ? w : 0.0f;
            tpe[e]  += kept;
            rppe[e] += kept * v[e];
            cum     += v[e];
        }
        entAcc += ent;
        wAcc   += w;
    }

    // stash per-thread 16-wide accumulators for the WMMA block reduction
#pragma unroll
    for (int e = 0; e < 16; ++e) {
        s_tpe [tid * 16 + e] = tpe[e];
        s_rppe[tid * 16 + e] = rppe[e];
    }

    // wave32 shuffle reduction for scalar sums
    for (int off = 16; off > 0; off >>= 1) {
        entAcc += __shfl_down(entAcc, off, 32);
        wAcc   += __shfl_down(wAcc,   off, 32);
    }
    if (lane == 0) {
        atomicAdd(&ws[32], entAcc);
        atomicAdd(&ws[33], wAcc);
    }

    __syncthreads();

    // Wave 0: reduce 256x16 LDS arrays to 16-wide column sums with f32 WMMA.
    // C = ones(16x4) * B(4x16) + C  => every row of C = column sums of B,
    // invariant to the exact K-lane layout of B (A is all ones).
    if (tid < 32) {                // whole wave 0 => EXEC all ones for WMMA
        v2f a; a[0] = 1.0f; a[1] = 1.0f;
        v8f c1 = {}; v8f c2 = {};
        const int col  = tid & 15;
        const int hsel = tid >> 4;
#pragma unroll 4
        for (int t = 0; t < 64; ++t) {    // 64 tiles of 4 threads each
            v2f b1, b2;
            b1[0] = s_tpe [(t * 4 + hsel    ) * 16 + col];
            b1[1] = s_tpe [(t * 4 + 2 + hsel) * 16 + col];
            b2[0] = s_rppe[(t * 4 + hsel    ) * 16 + col];
            b2[1] = s_rppe[(t * 4 + 2 + hsel) * 16 + col];
            c1 = __builtin_amdgcn_wmma_f32_16x16x4_f32(
                     false, a, false, b1, (short)0, c1, false, false);
            c2 = __builtin_amdgcn_wmma_f32_16x16x4_f32(
                     false, a, false, b2, (short)0, c2, false, false);
        }
        // C VGPR0, lanes 0..15 = row M=0, N=lane: the column sums
        if (tid < 16) {
            atomicAdd(&ws[col],      c1[0]);
            atomicAdd(&ws[16 + col], c2[0]);
        }
    }
}

__global__ void moe_router_finalize(const float* __restrict__ ws,
                                    float* __restrict__ out, float invN) {
    int lane = threadIdx.x;
    float prod = (lane < 16) ? ws[lane] * ws[16 + lane] : 0.0f;
    for (int off = 8; off > 0; off >>= 1) prod += __shfl_down(prod, off, 32);
    if (lane == 0) {
        float denom = ws[33];                       // = L * sum(attn)
        float inv2  = 1.0f / (denom * denom);
        float entLoss = ws[32] * invN;
        out[0] = entLoss * DYN_COEF + prod * inv2 * (float)E_EXP * AUX_COEF;
    }
}

extern "C" void kernel_launch(void* const* d_in, const int* in_sizes, int n_in,
                              void* d_out, int out_size, void* d_ws, size_t ws_size,
                              hipStream_t stream) {
    const float* logits = (const float*)d_in[0];
    const int*   attn   = (const int*)d_in[1];
    float*       ws     = (float*)d_ws;
    float*       out    = (float*)d_out;

    const int nTokens  = in_sizes[0] / E_EXP;   // L*B*S
    const int maskSize = in_sizes[1];           // B*S

    zero_ws_kernel<<<1, 64, 0, stream>>>(ws);

    int blocks = 1024;
    int maxBlocks = (nTokens + 255) / 256;
    if (blocks > maxBlocks) blocks = maxBlocks;
    if (blocks < 1) blocks = 1;

    moe_router_main<<<blocks, 256, 0, stream>>>(logits, attn, ws, nTokens, maskSize);
    moe_router_finalize<<<1, 32, 0, stream>>>(ws, out, 1.0f / (float)nTokens);
}